// SCELoss_70901320123271
// MI455X (gfx1250) — compile-verified
//
#include <hip/hip_runtime.h>
#include <hip/hip_bf16.h>
#include <cstdint>

// Problem geometry (fixed by the reference)
#define NROWS   4096
#define NCOLS   50001
#define NCLS    50000                      // last column dropped
#define BLOCK   1024                       // 32 wave32 waves
#define GRID    1024                       // blocks; each handles 4 rows
#define RPB     (NROWS / GRID)             // rows per block = 4
#define NCHUNK  ((NCLS + BLOCK - 1) / BLOCK)  // 49 async b32 loads per thread
#define EPSF        1e-7f
#define NEG_LOG_LOW 9.210340371976184f     // -log(1e-4)
#define SMEM_BYTES  ((NCLS + 64) * sizeof(float))  // row buffer + reduction scratch

// ---------------- wave32 / block reductions ----------------
__device__ __forceinline__ float waveMax(float v) {
#pragma unroll
  for (int o = 16; o > 0; o >>= 1) v = fmaxf(v, __shfl_xor(v, o, 32));
  return v;
}
__device__ __forceinline__ float waveSum(float v) {
#pragma unroll
  for (int o = 16; o > 0; o >>= 1) v += __shfl_xor(v, o, 32);
  return v;
}
// scr must have >= 33 floats. Leading barrier protects previous use of scr[32].
__device__ __forceinline__ float blockReduceMax(float v, float* scr) {
  const int lane = (int)threadIdx.x & 31, wid = (int)threadIdx.x >> 5;
  v = waveMax(v);
  __syncthreads();
  if (lane == 0) scr[wid] = v;
  __syncthreads();
  if (wid == 0) {
    float r = waveMax(scr[lane]);
    if (lane == 0) scr[32] = r;
  }
  __syncthreads();
  return scr[32];
}
__device__ __forceinline__ float blockReduceSum(float v, float* scr) {
  const int lane = (int)threadIdx.x & 31, wid = (int)threadIdx.x >> 5;
  v = waveSum(v);
  __syncthreads();
  if (lane == 0) scr[wid] = v;
  __syncthreads();
  if (wid == 0) {
    float r = waveSum(scr[lane]);
    if (lane == 0) scr[32] = r;
  }
  __syncthreads();
  return scr[32];
}

// ---------------- per-row kernel ----------------
// parts[0..NROWS)        = -log p_true  (CE contribution)
// parts[NROWS..2*NROWS)  = rce contribution
__global__ __launch_bounds__(BLOCK, 1)
void sce_row_kernel(const float* __restrict__ pred,
                    const long long* __restrict__ labels,
                    float* __restrict__ parts) {
  extern __shared__ float lds[];
  float* rowBuf = lds;            // NCLS floats, staged row
  float* scr    = lds + NCLS;     // [0..32] reduction scratch, [33] = true logit
  const int tid = (int)threadIdx.x;
  const uint64_t base = (uint64_t)(uintptr_t)pred;           // uniform -> SGPR pair
  const unsigned ldsBase = (unsigned)(uintptr_t)rowBuf;      // group-segment byte offset

  for (int rr = 0; rr < RPB; ++rr) {
    const int row = (int)blockIdx.x + rr * GRID;
    __syncthreads();  // previous row's LDS fully consumed before overwrite

    // ---- issue async global->LDS copy of the row (CDNA5 async-tensor path) ----
    // Uniform 49 issues/thread (tail clamped) so per-wave ASYNCcnt is exact.
    const unsigned rowByte = (unsigned)row * (unsigned)NCOLS * 4u;  // < 2^31
    for (int c = 0; c < NCHUNK; ++c) {
      int i = c * BLOCK + tid;
      if (i >= NCLS) i = NCLS - 1;
      const unsigned voff = rowByte + ((unsigned)i << 2);
      const unsigned ldsa = ldsBase + ((unsigned)i << 2);
      asm volatile("global_load_async_to_lds_b32 %0, %1, %2"
                   :: "v"(ldsa), "v"(voff), "s"(base)
                   : "memory");
    }

    // ---- pass 1: pipelined row max, overlapped with the in-flight copy ----
    // Each thread reads only slots it loaded; async loads complete in order
    // per wave, so a counted wait is sufficient (no barriers in this loop).
    float m = -3.402823466e38f;
#pragma unroll
    for (int c = 0; c < NCHUNK; ++c) {
      asm volatile("s_wait_asynccnt %0" :: "n"(NCHUNK - 1 - c) : "memory");
      int i = c * BLOCK + tid;
      if (i >= NCLS) i = NCLS - 1;
      m = fmaxf(m, rowBuf[i]);
    }
    const float M = blockReduceMax(m, scr);  // internal barriers => all waves at asynccnt==0

    // grab the true-class logit before pass 2 overwrites the buffer
    const int lab = (int)labels[row];
    if (tid == 0) scr[33] = rowBuf[lab];
    __syncthreads();

    // ---- prefetch this block's NEXT row into L2 while HBM is otherwise idle ----
    if (rr + 1 < RPB) {
      const char* nb = (const char*)pred + (size_t)(row + GRID) * NCOLS * 4;
      for (int off = tid * 128; off < NCLS * 4; off += BLOCK * 128)
        __builtin_prefetch(nb + off, 0, 1);   // -> global_prefetch_b8
    }

    // ---- pass 2: e = exp(x - M); stash e in place; accumulate sum ----
    float s = 0.0f;
    for (int i = tid; i < NCLS; i += BLOCK) {
      const float e = __expf(rowBuf[i] - M);  // v_exp_f32
      s += e;
      rowBuf[i] = e;
    }
    const float S = blockReduceSum(s, scr);

    // ---- pass 3: sum of clipped probabilities ----
    const float invS = 1.0f / S;
    float csum = 0.0f;
    for (int i = tid; i < NCLS; i += BLOCK) {
      float p = rowBuf[i] * invS;
      p = fminf(fmaxf(p, EPSF), 1.0f);
      csum += p;
    }
    const float C = blockReduceSum(csum, scr);

    if (tid == 0) {
      const float xt   = scr[33];
      const float logZ = M + __logf(S);
      const float lpt  = xt - logZ;                 // log p_true
      float pt = __expf(lpt);
      pt = fminf(fmaxf(pt, EPSF), 1.0f);            // clipped p_true
      parts[row]         = -lpt;                    // CE part
      parts[NROWS + row] = NEG_LOG_LOW * (C - pt);  // RCE part
    }
  }
}

// ---------------- deterministic final mean ----------------
__global__ __launch_bounds__(BLOCK, 1)
void sce_finalize(const float* __restrict__ parts, float* __restrict__ out) {
  __shared__ float scr[34];
  float a = 0.0f;
  for (int i = (int)threadIdx.x; i < 2 * NROWS; i += BLOCK) a += parts[i];
  const float t = blockReduceSum(a, scr);
  if (threadIdx.x == 0) out[0] = t * (1.0f / (float)NROWS);  // mean(ce)+mean(rce)
}

extern "C" void kernel_launch(void* const* d_in, const int* in_sizes, int n_in,
                              void* d_out, int out_size, void* d_ws, size_t ws_size,
                              hipStream_t stream) {
  (void)in_sizes; (void)n_in; (void)out_size; (void)ws_size;
  const float*     pred   = (const float*)d_in[0];
  const long long* labels = (const long long*)d_in[1];
  float*           parts  = (float*)d_ws;   // 2*NROWS floats = 32 KB scratch

  // 200 KB dynamic LDS per workgroup (<= 320 KB per WGP on CDNA5)
  hipFuncSetAttribute((const void*)sce_row_kernel,
                      hipFuncAttributeMaxDynamicSharedMemorySize, (int)SMEM_BYTES);

  sce_row_kernel<<<GRID, BLOCK, SMEM_BYTES, stream>>>(pred, labels, parts);
  sce_finalize<<<1, BLOCK, 0, stream>>>(parts, (float*)d_out);
}